// GraphClassifier_68899865363005
// MI455X (gfx1250) — compile-verified
//
#include <hip/hip_runtime.h>
#include <hip/hip_bf16.h>

#define NN 50000
#define EE 600000
#define FH 128
#define CC 10
#define GG 512
#define EPSV 1e-5f

typedef __attribute__((ext_vector_type(2))) float v2f;
typedef __attribute__((ext_vector_type(8))) float v8f;

// ---------------- utility ----------------
__global__ void k_fill(float* __restrict__ p, float v, int n) {
  int i = blockIdx.x * blockDim.x + threadIdx.x;
  if (i < n) p[i] = v;
}

// deg[dst] += 1 over edges (deg pre-filled with 1.0 for self loops)
__global__ void k_degree(const int* __restrict__ dst, float* __restrict__ deg, int e) {
  int i = blockIdx.x * blockDim.x + threadIdx.x;
  if (i < e) atomicAdd(&deg[dst[i]], 1.0f);
}

// in-place deg -> dinv = rsqrt(deg)  (deg >= 1 always)
__global__ void k_dinv(float* __restrict__ d, int n) {
  int i = blockIdx.x * blockDim.x + threadIdx.x;
  if (i < n) { float v = d[i]; d[i] = (v > 0.f) ? rsqrtf(v) : 0.f; }
}

// ---------------- WMMA f32 GEMM: Out[n][j] = sum_k In[n][k] * W[j][k] ----------------
// grid.x = Nrows/16 (row tiles), block = 256 = 8 waves; wave w -> output col tile j0=16w.
// Uses V_WMMA_F32_16X16X4_F32: A 16x4 (2 VGPR), B 4x16 (2 VGPR), C/D 16x16 f32 (8 VGPR).
__global__ void k_gemm_wmma(const float* __restrict__ In, const float* __restrict__ W,
                            float* __restrict__ Out, int Nrows) {
  __shared__ float xs[16 * FH]; // 8 KB X tile
  const int row0 = blockIdx.x * 16;
  const int tid  = threadIdx.x;
  for (int i = tid; i < 16 * FH; i += 256) {
    int r = i >> 7, c = i & 127;
    xs[i] = In[(size_t)(row0 + r) * FH + c];
  }
  __syncthreads();

  const int wave  = tid >> 5;          // 0..7
  const int lane  = tid & 31;
  const int j0    = wave * 16;         // output column tile
  const int m     = lane & 15;         // M (for A) / N (for B) index
  const int khalf = (lane >> 4) * 2;   // lanes 0-15: K+0/K+1 ; lanes 16-31: K+2/K+3

  v8f acc = {};
#pragma unroll
  for (int k = 0; k < FH; k += 4) {
    v2f a, b;
    // A-matrix 16x4 layout (ISA 7.12.2): lane<16 -> K=k+0,k+1 ; lane>=16 -> K=k+2,k+3
    a.x = xs[m * FH + k + khalf + 0];
    a.y = xs[m * FH + k + khalf + 1];
    // B-matrix 4x16: B[k'][j] = W[j0+j][k'] ; row K striped across lanes within VGPR
    b.x = W[(size_t)(j0 + m) * FH + k + khalf + 0];
    b.y = W[(size_t)(j0 + m) * FH + k + khalf + 1];
    acc = __builtin_amdgcn_wmma_f32_16x16x4_f32(
        /*neg_a=*/false, a, /*neg_b=*/false, b,
        /*c_mod=*/(short)0, acc, /*reuse_a=*/false, /*reuse_b=*/false);
  }

  // C/D layout: VGPR v -> M = v (+8 for lanes 16-31), N = lane%16
  const int nc    = lane & 15;
  const int mbase = (lane >> 4) * 8;
#pragma unroll
  for (int v = 0; v < 8; ++v) {
    Out[(size_t)(row0 + mbase + v) * FH + j0 + nc] = acc[v];
  }
}

// ---------------- GCN aggregation ----------------
// out[n][f] = bias[f] + dinv[n]^2 * h[n][f]   (self-loop term, initializes out)
__global__ void k_self_init(const float* __restrict__ h, const float* __restrict__ dinv,
                            const float* __restrict__ bias, float* __restrict__ out, int n) {
  int i = blockIdx.x * blockDim.x + threadIdx.x;
  if (i < n * FH) {
    int node = i >> 7, f = i & 127;
    float d = dinv[node];
    out[i] = bias[f] + d * d * h[i];
  }
}

// out[dst][f] += dinv[src]*dinv[dst] * h[src][f]  — thread per (edge, 4-feature group)
__global__ void k_scatter(const int* __restrict__ src, const int* __restrict__ dst,
                          const float* __restrict__ dinv, const float* __restrict__ h,
                          float* __restrict__ out, int e) {
  int t = blockIdx.x * blockDim.x + threadIdx.x;
  if (t >= e * 32) return;
  int ed = t >> 5;
  int f4 = (t & 31) << 2;
  int s = src[ed], d = dst[ed];
  float nrm = dinv[s] * dinv[d];
  const float4 hv = *(const float4*)(h + (size_t)s * FH + f4);
  float* o = out + (size_t)d * FH + f4;
  atomicAdd(o + 0, nrm * hv.x);
  atomicAdd(o + 1, nrm * hv.y);
  atomicAdd(o + 2, nrm * hv.z);
  atomicAdd(o + 3, nrm * hv.w);
}

// ---------------- BatchNorm (affine=False, training stats) + ReLU ----------------
// stats[0:128]=sum, stats[128:256]=sumsq ; block = 128 threads (one per feature), 64 rows/block
__global__ void k_bn_stats(const float* __restrict__ x, float* __restrict__ stats, int n) {
  int f  = threadIdx.x;
  int r0 = blockIdx.x * 64;
  int r1 = r0 + 64; if (r1 > n) r1 = n;
  float s = 0.f, s2 = 0.f;
  for (int r = r0; r < r1; ++r) {
    float v = x[(size_t)r * FH + f];
    s += v; s2 += v * v;
  }
  atomicAdd(&stats[f], s);
  atomicAdd(&stats[FH + f], s2);
}

__global__ void k_bn_relu(float* __restrict__ x, const float* __restrict__ stats, int n) {
  int i = blockIdx.x * blockDim.x + threadIdx.x;
  if (i < n * FH) {
    int f = i & 127;
    const float invN = 1.0f / (float)NN;
    float mu  = stats[f] * invN;
    float var = stats[FH + f] * invN - mu * mu;
    float y = (x[i] - mu) * rsqrtf(var + EPSV);
    x[i] = fmaxf(y, 0.0f);
  }
}

// ---------------- global mean pool + final linear ----------------
__global__ void k_pool(const float* __restrict__ h, const int* __restrict__ batch,
                       float* __restrict__ pooled, float* __restrict__ cnts, int n) {
  int t = blockIdx.x * blockDim.x + threadIdx.x;
  if (t >= n * 32) return;
  int node = t >> 5;
  int f4 = (t & 31) << 2;
  int g = batch[node];
  const float4 hv = *(const float4*)(h + (size_t)node * FH + f4);
  float* p = pooled + (size_t)g * FH + f4;
  atomicAdd(p + 0, hv.x);
  atomicAdd(p + 1, hv.y);
  atomicAdd(p + 2, hv.z);
  atomicAdd(p + 3, hv.w);
  if (f4 == 0) atomicAdd(&cnts[g], 1.0f);
}

__global__ void k_final(const float* __restrict__ pooled, const float* __restrict__ cnts,
                        const float* __restrict__ Wlin, const float* __restrict__ blin,
                        float* __restrict__ out) {
  int t = blockIdx.x * blockDim.x + threadIdx.x;
  if (t >= GG * CC) return;
  int g = t / CC, c = t % CC;
  float inv = 1.0f / fmaxf(cnts[g], 1.0f);
  float s = 0.f;
  for (int k = 0; k < FH; ++k)
    s += pooled[(size_t)g * FH + k] * Wlin[(size_t)c * FH + k];
  out[t] = s * inv + blin[c];
}

// ---------------- host launch ----------------
extern "C" void kernel_launch(void* const* d_in, const int* in_sizes, int n_in,
                              void* d_out, int out_size, void* d_ws, size_t ws_size,
                              hipStream_t stream) {
  (void)in_sizes; (void)n_in; (void)out_size; (void)ws_size;

  const float* x     = (const float*)d_in[0];
  const int*   eidx  = (const int*)d_in[1];   // [2, E] : row0 = src, row1 = dst
  const int*   batch = (const int*)d_in[2];
  const float* W1 = (const float*)d_in[3];  const float* b1 = (const float*)d_in[4];
  const float* W2 = (const float*)d_in[5];  const float* b2 = (const float*)d_in[6];
  const float* W3 = (const float*)d_in[7];  const float* b3 = (const float*)d_in[8];
  const float* Wl = (const float*)d_in[9];  const float* bl = (const float*)d_in[10];
  const int* src = eidx;
  const int* dst = eidx + EE;
  float* out = (float*)d_out;

  // workspace layout (floats, all offsets 16B-aligned)
  float* ws     = (float*)d_ws;
  float* dinv   = ws;                               // N   (deg -> dinv in place)
  float* T      = dinv + NN;                        // N*H (gemm output h)
  float* U      = T + (size_t)NN * FH;              // N*H (aggregated / activations)
  float* stats  = U + (size_t)NN * FH;              // 256
  float* pooled = stats + 256;                      // G*H
  float* cnts   = pooled + (size_t)GG * FH;         // G

  const int TB = 256;
  dim3 blk(TB);
  auto gridFor = [](long long n, int tb) { return dim3((unsigned)((n + tb - 1) / tb)); };

  // degrees / norm
  k_fill<<<gridFor(NN, TB), blk, 0, stream>>>(dinv, 1.0f, NN);
  k_degree<<<gridFor(EE, TB), blk, 0, stream>>>(dst, dinv, EE);
  k_dinv<<<gridFor(NN, TB), blk, 0, stream>>>(dinv, NN);

  const dim3 gemm_grid(NN / 16);     // 3125 row tiles
  const long long nf  = (long long)NN * FH;
  const long long e32 = (long long)EE * 32;

  // ---- layer 1 ----
  k_gemm_wmma<<<gemm_grid, blk, 0, stream>>>(x, W1, T, NN);
  k_self_init<<<gridFor(nf, TB), blk, 0, stream>>>(T, dinv, b1, U, NN);
  k_scatter<<<gridFor(e32, TB), blk, 0, stream>>>(src, dst, dinv, T, U, EE);
  k_fill<<<dim3(1), blk, 0, stream>>>(stats, 0.0f, 256);
  k_bn_stats<<<gridFor(NN, 64), dim3(FH), 0, stream>>>(U, stats, NN);
  k_bn_relu<<<gridFor(nf, TB), blk, 0, stream>>>(U, stats, NN);

  // ---- layer 2 ----
  k_gemm_wmma<<<gemm_grid, blk, 0, stream>>>(U, W2, T, NN);
  k_self_init<<<gridFor(nf, TB), blk, 0, stream>>>(T, dinv, b2, U, NN);
  k_scatter<<<gridFor(e32, TB), blk, 0, stream>>>(src, dst, dinv, T, U, EE);
  k_fill<<<dim3(1), blk, 0, stream>>>(stats, 0.0f, 256);
  k_bn_stats<<<gridFor(NN, 64), dim3(FH), 0, stream>>>(U, stats, NN);
  k_bn_relu<<<gridFor(nf, TB), blk, 0, stream>>>(U, stats, NN);

  // ---- layer 3 (no BN/ReLU) ----
  k_gemm_wmma<<<gemm_grid, blk, 0, stream>>>(U, W3, T, NN);
  k_self_init<<<gridFor(nf, TB), blk, 0, stream>>>(T, dinv, b3, U, NN);
  k_scatter<<<gridFor(e32, TB), blk, 0, stream>>>(src, dst, dinv, T, U, EE);

  // ---- pool + classifier head ----
  k_fill<<<gridFor(GG * FH, TB), blk, 0, stream>>>(pooled, 0.0f, GG * FH);
  k_fill<<<gridFor(GG, TB), blk, 0, stream>>>(cnts, 0.0f, GG);
  k_pool<<<gridFor((long long)NN * 32, TB), blk, 0, stream>>>(U, batch, pooled, cnts, NN);
  k_final<<<gridFor(GG * CC, TB), blk, 0, stream>>>(pooled, cnts, Wl, bl, out);
}